// GINNet_7713761263893
// MI455X (gfx1250) — compile-verified
//
#include <hip/hip_runtime.h>

#define NNODES 50000
#define NEDGES 800000
#define DIM    96
#define HDIM   32
#define OUTD   4

typedef float v2f __attribute__((ext_vector_type(2)));
typedef float v8f __attribute__((ext_vector_type(8)));

__device__ __forceinline__ v8f wmma_f32(v2f a, v2f b, v8f c) {
  // D = A(16x4) * B(4x16) + C(16x16), full fp32
  return __builtin_amdgcn_wmma_f32_16x16x4_f32(false, a, false, b, (short)0, c,
                                               false, false);
}

// z[i] = (1 + eps) * src[i], float4-vectorized
__global__ __launch_bounds__(256) void scale_kernel(const float4* __restrict__ src,
                                                    float4* __restrict__ dst,
                                                    const float* __restrict__ eps,
                                                    int n4) {
  int i = blockIdx.x * blockDim.x + threadIdx.x;
  if (i >= n4) return;
  float s = 1.0f + eps[0];
  float4 v = src[i];
  v.x *= s; v.y *= s; v.z *= s; v.w *= s;
  dst[i] = v;
}

// accum[dst[e]] += feat[src[e]] ; one thread per (edge, 4-feature chunk)
__global__ __launch_bounds__(256) void scatter_kernel(const float* __restrict__ feat,
                                                      const int* __restrict__ srcIdx,
                                                      const int* __restrict__ dstIdx,
                                                      float* __restrict__ accum) {
  long long t = (long long)blockIdx.x * blockDim.x + threadIdx.x;
  const long long total = (long long)NEDGES * (DIM / 4);
  if (t >= total) return;
  int e = (int)(t / (DIM / 4));
  int c = (int)(t % (DIM / 4));
  int s = srcIdx[e];
  int d = dstIdx[e];
  float4 v = *(const float4*)(feat + (size_t)s * DIM + c * 4);
  float* p = accum + (size_t)d * DIM + c * 4;
  unsafeAtomicAdd(p + 0, v.x);
  unsafeAtomicAdd(p + 1, v.y);
  unsafeAtomicAdd(p + 2, v.z);
  unsafeAtomicAdd(p + 3, v.w);
}

// h = relu( relu(z@W1a + b1a) @ W1b + b1b ), z:[N,96], W:[96,96]
// 128 threads = 4 waves; each wave owns a 16-node row tile (64 nodes / block).
__global__ __launch_bounds__(128) void mlp1_kernel(const float* __restrict__ z,
                                                   const float* __restrict__ W1a,
                                                   const float* __restrict__ b1a,
                                                   const float* __restrict__ W1b,
                                                   const float* __restrict__ b1b,
                                                   float* __restrict__ h) {
  __shared__ float sW[DIM * DIM];        // 36 KB, reused for W1a then W1b
  __shared__ float sT[4][16 * DIM];      // 24 KB, per-wave intermediate tiles

  const int tid  = threadIdx.x;
  const int lane = tid & 31;
  const int wave = tid >> 5;
  const int half = lane >> 4;            // 0: lanes 0-15, 1: lanes 16-31
  const int m    = lane & 15;            // row within tile (A) / col (B,D)
  const int n    = lane & 15;
  const int kk   = half * 2;             // K sub-offset for A/B fragments
  const int nodeBase = blockIdx.x * 64 + wave * 16;
  const int row  = nodeBase + m;
  const bool rowOk = row < NNODES;

  // ---- stage W1a ----
  for (int i = tid; i < DIM * DIM; i += 128) sW[i] = W1a[i];
  __syncthreads();

  // ---- A fragments for GEMM1 from global z ----
  v2f a[DIM / 4];
  for (int kt = 0; kt < DIM / 4; ++kt) {
    int k = kt * 4 + kk;
    if (rowOk) a[kt] = *(const v2f*)(z + (size_t)row * DIM + k);
    else       a[kt] = v2f{0.0f, 0.0f};
  }

  // ---- GEMM1: 6 N-tiles, K=96 ----
  for (int j = 0; j < DIM / 16; ++j) {
    int n0 = j * 16;
    v8f acc = {};
    for (int kt = 0; kt < DIM / 4; ++kt) {
      int k = kt * 4 + kk;
      v2f b;
      b.x = sW[(k)     * DIM + n0 + n];
      b.y = sW[(k + 1) * DIM + n0 + n];
      acc = wmma_f32(a[kt], b, acc);
    }
    float bias = b1a[n0 + n];
    for (int v = 0; v < 8; ++v) {
      float val = acc[v] + bias;
      val = val > 0.0f ? val : 0.0f;                     // inner ReLU
      int mm = v + half * 8;
      sT[wave][mm * DIM + n0 + n] = val;
    }
  }
  __syncthreads();

  // ---- stage W1b (reuse sW) ----
  for (int i = tid; i < DIM * DIM; i += 128) sW[i] = W1b[i];
  __syncthreads();

  // ---- A fragments for GEMM2 from LDS intermediate ----
  for (int kt = 0; kt < DIM / 4; ++kt) {
    int k = kt * 4 + kk;
    a[kt].x = sT[wave][m * DIM + k];
    a[kt].y = sT[wave][m * DIM + k + 1];
  }

  // ---- GEMM2 + outer ReLU -> h ----
  for (int j = 0; j < DIM / 16; ++j) {
    int n0 = j * 16;
    v8f acc = {};
    for (int kt = 0; kt < DIM / 4; ++kt) {
      int k = kt * 4 + kk;
      v2f b;
      b.x = sW[(k)     * DIM + n0 + n];
      b.y = sW[(k + 1) * DIM + n0 + n];
      acc = wmma_f32(a[kt], b, acc);
    }
    float bias = b1b[n0 + n];
    for (int v = 0; v < 8; ++v) {
      float val = acc[v] + bias;
      val = val > 0.0f ? val : 0.0f;                     // outer ReLU
      int mm = v + half * 8;
      int r = nodeBase + mm;
      if (r < NNODES) h[(size_t)r * DIM + n0 + n] = val;
    }
  }
}

// out = relu( relu(z2@W2a + b2a) @ W2b + b2b ) @ Wh + bh
__global__ __launch_bounds__(128) void mlp2_kernel(const float* __restrict__ z2,
                                                   const float* __restrict__ W2a,
                                                   const float* __restrict__ b2a,
                                                   const float* __restrict__ W2b,
                                                   const float* __restrict__ b2b,
                                                   const float* __restrict__ Wh,
                                                   const float* __restrict__ bh,
                                                   float* __restrict__ out) {
  __shared__ float sW2a[DIM * HDIM];     // 12 KB
  __shared__ float sW2b[HDIM * HDIM];    //  4 KB
  __shared__ float sWh[HDIM * OUTD];     // 0.5 KB
  __shared__ float sT[4][16 * HDIM];     //  8 KB
  __shared__ float sH2[4][16 * HDIM];    //  8 KB

  const int tid  = threadIdx.x;
  const int lane = tid & 31;
  const int wave = tid >> 5;
  const int half = lane >> 4;
  const int m    = lane & 15;
  const int n    = lane & 15;
  const int kk   = half * 2;
  const int nodeBase = blockIdx.x * 64 + wave * 16;
  const int row  = nodeBase + m;
  const bool rowOk = row < NNODES;

  for (int i = tid; i < DIM * HDIM;  i += 128) sW2a[i] = W2a[i];
  for (int i = tid; i < HDIM * HDIM; i += 128) sW2b[i] = W2b[i];
  for (int i = tid; i < HDIM * OUTD; i += 128) sWh[i]  = Wh[i];
  __syncthreads();

  // ---- GEMM A: z2[16x96] @ W2a[96x32] ----
  v2f a[DIM / 4];
  for (int kt = 0; kt < DIM / 4; ++kt) {
    int k = kt * 4 + kk;
    if (rowOk) a[kt] = *(const v2f*)(z2 + (size_t)row * DIM + k);
    else       a[kt] = v2f{0.0f, 0.0f};
  }
  for (int j = 0; j < HDIM / 16; ++j) {
    int n0 = j * 16;
    v8f acc = {};
    for (int kt = 0; kt < DIM / 4; ++kt) {
      int k = kt * 4 + kk;
      v2f b;
      b.x = sW2a[(k)     * HDIM + n0 + n];
      b.y = sW2a[(k + 1) * HDIM + n0 + n];
      acc = wmma_f32(a[kt], b, acc);
    }
    float bias = b2a[n0 + n];
    for (int v = 0; v < 8; ++v) {
      float val = acc[v] + bias;
      val = val > 0.0f ? val : 0.0f;
      int mm = v + half * 8;
      sT[wave][mm * HDIM + n0 + n] = val;
    }
  }
  __syncthreads();

  // ---- GEMM B: sT[16x32] @ W2b[32x32], outer ReLU ----
  v2f a2[HDIM / 4];
  for (int kt = 0; kt < HDIM / 4; ++kt) {
    int k = kt * 4 + kk;
    a2[kt].x = sT[wave][m * HDIM + k];
    a2[kt].y = sT[wave][m * HDIM + k + 1];
  }
  for (int j = 0; j < HDIM / 16; ++j) {
    int n0 = j * 16;
    v8f acc = {};
    for (int kt = 0; kt < HDIM / 4; ++kt) {
      int k = kt * 4 + kk;
      v2f b;
      b.x = sW2b[(k)     * HDIM + n0 + n];
      b.y = sW2b[(k + 1) * HDIM + n0 + n];
      acc = wmma_f32(a2[kt], b, acc);
    }
    float bias = b2b[n0 + n];
    for (int v = 0; v < 8; ++v) {
      float val = acc[v] + bias;
      val = val > 0.0f ? val : 0.0f;
      int mm = v + half * 8;
      sH2[wave][mm * HDIM + n0 + n] = val;
    }
  }
  __syncthreads();

  // ---- Head: h2[16x32] @ Wh[32x4] + bh ; lanes 0-15 -> outs 0,1; 16-31 -> 2,3
  for (int oo = 0; oo < 2; ++oo) {
    int o = half * 2 + oo;
    float accv = bh[o];
    for (int k = 0; k < HDIM; ++k)
      accv += sH2[wave][m * HDIM + k] * sWh[k * OUTD + o];
    if (rowOk) out[(size_t)row * OUTD + o] = accv;
  }
}

extern "C" void kernel_launch(void* const* d_in, const int* in_sizes, int n_in,
                              void* d_out, int out_size, void* d_ws, size_t ws_size,
                              hipStream_t stream) {
  const float* x    = (const float*)d_in[0];
  const int*   ei   = (const int*)d_in[1];   // [2, E]: src then dst
  const float* eps1 = (const float*)d_in[2];
  const float* eps2 = (const float*)d_in[3];
  const float* W1a  = (const float*)d_in[4];
  const float* b1a  = (const float*)d_in[5];
  const float* W1b  = (const float*)d_in[6];
  const float* b1b  = (const float*)d_in[7];
  const float* W2a  = (const float*)d_in[8];
  const float* b2a  = (const float*)d_in[9];
  const float* W2b  = (const float*)d_in[10];
  const float* b2b  = (const float*)d_in[11];
  const float* Wh   = (const float*)d_in[12];
  const float* bh   = (const float*)d_in[13];

  float* z = (float*)d_ws;                       // [N, 96] agg buffer (reused)
  float* h = z + (size_t)NNODES * DIM;           // [N, 96] layer-1 output

  const int n4 = NNODES * DIM / 4;
  const int scaleBlocks = (n4 + 255) / 256;
  const long long sThreads = (long long)NEDGES * (DIM / 4);
  const int scatBlocks = (int)((sThreads + 255) / 256);
  const int mlpBlocks = (NNODES + 63) / 64;

  // Layer 1
  scale_kernel<<<scaleBlocks, 256, 0, stream>>>((const float4*)x, (float4*)z, eps1, n4);
  scatter_kernel<<<scatBlocks, 256, 0, stream>>>(x, ei, ei + NEDGES, z);
  mlp1_kernel<<<mlpBlocks, 128, 0, stream>>>(z, W1a, b1a, W1b, b1b, h);

  // Layer 2 (+head), reusing z as z2
  scale_kernel<<<scaleBlocks, 256, 0, stream>>>((const float4*)h, (float4*)z, eps2, n4);
  scatter_kernel<<<scatBlocks, 256, 0, stream>>>(h, ei, ei + NEDGES, z);
  mlp2_kernel<<<mlpBlocks, 128, 0, stream>>>(z, W2a, b2a, W2b, b2b, Wh, bh, (float*)d_out);
}